// DigitCapsuleLayer_42829413875977
// MI455X (gfx1250) — compile-verified
//
#include <hip/hip_runtime.h>
#include <hip/hip_bf16.h>
#include <math.h>

// Problem constants
#define NB   256    // batch
#define NR   1152   // routes
#define NC   10     // capsules
#define NO   16     // output dim per capsule
#define NI   8      // input dim per capsule
#define NG   32     // distinct W groups (route r uses W[r % 32])

typedef float v2f __attribute__((ext_vector_type(2)));
typedef float v8f __attribute__((ext_vector_type(8)));

// ---------------------------------------------------------------------------
// c_ij = softmax(b_ij, axis=1)   (per-route softmax over 10 capsules)
// ---------------------------------------------------------------------------
__global__ void caps_softmax_kernel(const float* __restrict__ bij,
                                    float* __restrict__ cij) {
    int r = blockIdx.x * blockDim.x + threadIdx.x;
    if (r >= NR) return;
    float x[NC];
    float m = -INFINITY;
#pragma unroll
    for (int j = 0; j < NC; ++j) {
        x[j] = bij[r * NC + j];
        m = fmaxf(m, x[j]);
    }
    float sum = 0.0f;
#pragma unroll
    for (int j = 0; j < NC; ++j) {
        x[j] = expf(x[j] - m);
        sum += x[j];
    }
    float inv = 1.0f / sum;
#pragma unroll
    for (int j = 0; j < NC; ++j) cij[r * NC + j] = x[j] * inv;
}

// ---------------------------------------------------------------------------
// s[b,c,o] += sum_{r in chunk} c_ij[r,c] * (W[r%32][c] @ u[b,r])[o]
// Block: 320 threads = 10 waves, wave w handles capsule c = w.
// Grid: (batch tiles of 16, route chunks). u_hat never materialized:
// 2x v_wmma_f32_16x16x4_f32 per route per capsule, scale folded into B.
// ---------------------------------------------------------------------------
__global__ void caps_s_kernel(const float* __restrict__ u,
                              const float* __restrict__ W,
                              const float* __restrict__ cij,
                              float* __restrict__ s,
                              int routes_per_block) {
    const int lane = threadIdx.x & 31;
    const int c    = threadIdx.x >> 5;            // capsule handled by this wave
    const int bt0  = blockIdx.x * 16;             // batch-tile base
    const int r0   = blockIdx.y * routes_per_block;
    const int r1   = r0 + routes_per_block;

    const int mrow  = lane & 15;                  // M index (A) / N index (B)
    const int khalf = (lane >> 4) * 2;            // +0 for lanes 0-15, +2 for 16-31

    v8f acc = {0.f, 0.f, 0.f, 0.f, 0.f, 0.f, 0.f, 0.f};

    for (int r = r0; r < r1; ++r) {
        const int g = r & (NG - 1);
        const float scale = cij[r * NC + c];

        // A fragment: u[bt0+mrow, r, k]  (k pairs {khalf,khalf+1} and +4)
        const float* up = u + ((size_t)(bt0 + mrow) * NR + (size_t)r) * NI + khalf;
        v2f a0 = *(const v2f*)(up);
        v2f a1 = *(const v2f*)(up + 4);

        // B fragment: W[g, c, o=mrow, k] * c_ij[r,c]
        const float* wp = W + (((size_t)g * NC + c) * NO + mrow) * NI + khalf;
        v2f bf0 = *(const v2f*)(wp);
        v2f bf1 = *(const v2f*)(wp + 4);
        bf0 *= scale;
        bf1 *= scale;

        acc = __builtin_amdgcn_wmma_f32_16x16x4_f32(false, a0, false, bf0,
                                                    (short)0, acc, false, false);
        acc = __builtin_amdgcn_wmma_f32_16x16x4_f32(false, a1, false, bf1,
                                                    (short)0, acc, false, false);
    }

    // C/D layout: VGPR k -> (M = k or k+8 by lane half, N = mrow)
    const int o     = mrow;
    const int bbase = bt0 + (lane >> 4) * 8;
#pragma unroll
    for (int k = 0; k < 8; ++k) {
        atomicAdd(&s[(size_t)(bbase + k) * (NC * NO) + c * NO + o], acc[k]);
    }
}

// ---------------------------------------------------------------------------
// v = squash(s): mag_sq/(1+mag_sq) * s/|s| per (b,c) 16-vector.
// ---------------------------------------------------------------------------
__global__ void caps_squash_kernel(const float* __restrict__ s,
                                   float* __restrict__ v,
                                   float* __restrict__ out,
                                   int write_out) {
    int idx = blockIdx.x * blockDim.x + threadIdx.x;   // (b,c) pair
    if (idx >= NB * NC) return;
    const float* sp = s + (size_t)idx * NO;
    float t[NO];
    float ms = 0.0f;
#pragma unroll
    for (int o = 0; o < NO; ++o) {
        t[o] = sp[o];
        ms += t[o] * t[o];
    }
    float f = (ms / (1.0f + ms)) / sqrtf(ms);
#pragma unroll
    for (int o = 0; o < NO; ++o) {
        float val = t[o] * f;
        v[(size_t)idx * NO + o] = val;
        if (write_out) out[(size_t)idx * NO + o] = val;
    }
}

// ---------------------------------------------------------------------------
// b_ij[r,c] += (1/B) * sum_{b,o} u_hat[b,r,c,o] * v[b,c,o]
// Same WMMA tiling as s-kernel (C=0 recompute of the u_hat tile), v tile
// hoisted outside the route loop, wave32 shfl_xor reduction, lane-0 atomic.
// ---------------------------------------------------------------------------
__global__ void caps_a_kernel(const float* __restrict__ u,
                              const float* __restrict__ W,
                              const float* __restrict__ v,
                              float* __restrict__ bij,
                              int routes_per_block) {
    const int lane = threadIdx.x & 31;
    const int c    = threadIdx.x >> 5;
    const int bt0  = blockIdx.x * 16;
    const int r0   = blockIdx.y * routes_per_block;
    const int r1   = r0 + routes_per_block;

    const int mrow  = lane & 15;
    const int khalf = (lane >> 4) * 2;

    // Hoist v tile in C/D fragment layout (invariant over routes)
    const int o     = mrow;
    const int bbase = bt0 + (lane >> 4) * 8;
    float vf[8];
#pragma unroll
    for (int k = 0; k < 8; ++k) {
        vf[k] = v[(size_t)(bbase + k) * (NC * NO) + c * NO + o];
    }

    const float invB = 1.0f / (float)NB;

    for (int r = r0; r < r1; ++r) {
        const int g = r & (NG - 1);

        const float* up = u + ((size_t)(bt0 + mrow) * NR + (size_t)r) * NI + khalf;
        v2f a0 = *(const v2f*)(up);
        v2f a1 = *(const v2f*)(up + 4);

        const float* wp = W + (((size_t)g * NC + c) * NO + mrow) * NI + khalf;
        v2f bf0 = *(const v2f*)(wp);
        v2f bf1 = *(const v2f*)(wp + 4);

        v8f acc = {0.f, 0.f, 0.f, 0.f, 0.f, 0.f, 0.f, 0.f};
        acc = __builtin_amdgcn_wmma_f32_16x16x4_f32(false, a0, false, bf0,
                                                    (short)0, acc, false, false);
        acc = __builtin_amdgcn_wmma_f32_16x16x4_f32(false, a1, false, bf1,
                                                    (short)0, acc, false, false);

        // <u_hat tile, v tile> over this wave's 16x16 tile
        float dot = 0.0f;
#pragma unroll
        for (int k = 0; k < 8; ++k) dot += acc[k] * vf[k];
        // wave32 reduction
#pragma unroll
        for (int off = 16; off > 0; off >>= 1) dot += __shfl_xor(dot, off, 32);

        if (lane == 0) atomicAdd(&bij[r * NC + c], dot * invB);
    }
}

// ---------------------------------------------------------------------------
// Launch: 3 routing iterations, global sync via kernel boundaries.
// ---------------------------------------------------------------------------
extern "C" void kernel_launch(void* const* d_in, const int* in_sizes, int n_in,
                              void* d_out, int out_size, void* d_ws, size_t ws_size,
                              hipStream_t stream) {
    const float* u = (const float*)d_in[0];   // [256,1152,8]
    const float* W = (const float*)d_in[1];   // [32,10,16,8]
    float* out = (float*)d_out;               // [256,10,16,1]

    float* ws  = (float*)d_ws;
    float* bij = ws;                          // 1152*10
    float* cij = bij + NR * NC;               // 1152*10
    float* s   = cij + NR * NC;               // 256*10*16
    float* v   = s + NB * NC * NO;            // 256*10*16

    hipMemsetAsync(bij, 0, (size_t)NR * NC * sizeof(float), stream);

    const int ROUTES_PER_BLOCK = 128;         // 1152/128 = 9 chunks
    dim3 mmGrid(NB / 16, NR / ROUTES_PER_BLOCK);
    dim3 mmBlock(NC * 32);                    // 10 waves, one capsule each

    for (int it = 0; it < 3; ++it) {
        caps_softmax_kernel<<<dim3((NR + 127) / 128), dim3(128), 0, stream>>>(bij, cij);
        hipMemsetAsync(s, 0, (size_t)NB * NC * NO * sizeof(float), stream);
        caps_s_kernel<<<mmGrid, mmBlock, 0, stream>>>(u, W, cij, s, ROUTES_PER_BLOCK);
        caps_squash_kernel<<<dim3((NB * NC + 255) / 256), dim3(256), 0, stream>>>(
            s, v, out, (it == 2) ? 1 : 0);
        if (it < 2) {
            caps_a_kernel<<<mmGrid, mmBlock, 0, stream>>>(u, W, v, bij, ROUTES_PER_BLOCK);
        }
    }
}